// CONT_Loss_13649406066960
// MI455X (gfx1250) — compile-verified
//
#include <hip/hip_runtime.h>
#include <hip/hip_bf16.h>
#include <math.h>

typedef __attribute__((ext_vector_type(2))) float v2f;
typedef __attribute__((ext_vector_type(8))) float v8f;

#define N_TOTAL 8192      // B*C = 512*16
#define D_FEAT  128
#define N_TILES 512       // N_TOTAL / 16
#define MASKED_LOGIT (-1000000.0f)   // MASK_VAL / TEMPERATURE

// One wave (32 threads) per 16-row tile. Computes row-wise logsumexp of
// logits = 100 * (R @ Cm^T) with block-diagonal masking (tile j==r:
// off-diagonal entries -> -1e6, diagonal kept), streaming over all 512
// column tiles with V_WMMA_F32_16X16X4_F32.
__global__ __launch_bounds__(32)
void lse_rows_kernel(const float* __restrict__ R,
                     const float* __restrict__ Cm,
                     float* __restrict__ lse_out) {
    const int lane = threadIdx.x & 31;
    const int r    = blockIdx.x;          // 16-row tile index
    const int mloc = lane & 15;           // A: row M   / B: col N
    const int koff = (lane >> 4) * 2;     // upper half handles K+2, K+3

    // A fragments: lane holds rows m = mloc, elements (K = 4t+koff, 4t+koff+1)
    v2f a[32];
    const float* arow = R + (size_t)(r * 16 + mloc) * D_FEAT + koff;
#pragma unroll
    for (int t = 0; t < 32; ++t)
        a[t] = *(const v2f*)(arow + 4 * t);

    // Per-lane online logsumexp state for the 8 row slots this lane owns.
    float mrun[8], srun[8];
#pragma unroll
    for (int v = 0; v < 8; ++v) { mrun[v] = -INFINITY; srun[v] = 0.0f; }

    for (int j = 0; j < N_TILES; ++j) {
        const float* brow = Cm + (size_t)(j * 16 + mloc) * D_FEAT + koff;
        v8f c0 = {};
        v8f c1 = {};
        // K = 128 as 32 steps of K=4; two accumulator chains.
#pragma unroll
        for (int t = 0; t < 32; t += 2) {
            v2f b0 = *(const v2f*)(brow + 4 * t);
            v2f b1 = *(const v2f*)(brow + 4 * (t + 1));
            c0 = __builtin_amdgcn_wmma_f32_16x16x4_f32(
                     false, a[t],     false, b0, (short)0, c0, false, false);
            c1 = __builtin_amdgcn_wmma_f32_16x16x4_f32(
                     false, a[t + 1], false, b1, (short)0, c1, false, false);
        }
        const bool diagTile = (j == r);   // wave-uniform
#pragma unroll
        for (int v = 0; v < 8; ++v) {
            float logit = 100.0f * (c0[v] + c1[v]);
            if (diagTile) {
                int Mg = v + ((lane >> 4) << 3);   // local row in tile
                if (Mg != mloc) logit = MASKED_LOGIT;
            }
            float nm = fmaxf(mrun[v], logit);
            srun[v] = srun[v] * __expf(mrun[v] - nm) + __expf(logit - nm);
            mrun[v] = nm;
        }
    }

    // Merge (max,sum) across the 16 lanes sharing each row (within halves).
#pragma unroll
    for (int off = 1; off < 16; off <<= 1) {
#pragma unroll
        for (int v = 0; v < 8; ++v) {
            float m2 = __shfl_xor(mrun[v], off, 32);
            float s2 = __shfl_xor(srun[v], off, 32);
            float nm = fmaxf(mrun[v], m2);
            srun[v] = srun[v] * __expf(mrun[v] - nm) + s2 * __expf(m2 - nm);
            mrun[v] = nm;
        }
    }

    if (lane == 0 || lane == 16) {
        int base = r * 16 + ((lane >> 4) << 3);
#pragma unroll
        for (int v = 0; v < 8; ++v)
            lse_out[base + v] = mrun[v] + __logf(srun[v]);
    }
}

// Single-block finalize: diagonal logits + mean + nan/inf guard.
__global__ __launch_bounds__(256)
void finalize_kernel(const float* __restrict__ ts,
                     const float* __restrict__ nt,
                     const float* __restrict__ lse_r,
                     const float* __restrict__ lse_c,
                     float* __restrict__ out) {
    __shared__ float red[256];
    const int t = threadIdx.x;
    float acc = 0.0f;
    for (int i = t; i < N_TOTAL; i += 256) {
        const float4* av = (const float4*)(ts + (size_t)i * D_FEAT);
        const float4* bv = (const float4*)(nt + (size_t)i * D_FEAT);
        float d = 0.0f;
#pragma unroll
        for (int q = 0; q < D_FEAT / 4; ++q) {
            float4 x = av[q], y = bv[q];
            d += x.x * y.x + x.y * y.y + x.z * y.z + x.w * y.w;
        }
        acc += 0.5f * (lse_r[i] + lse_c[i]) - 100.0f * d;
    }
    red[t] = acc;
    __syncthreads();
    for (int s = 128; s > 0; s >>= 1) {
        if (t < s) red[t] += red[t + s];
        __syncthreads();
    }
    if (t == 0) {
        float v = red[0] / (float)N_TOTAL;
        if (isnan(v) || isinf(v)) v = 0.0f;
        out[0] = v;
    }
}

extern "C" void kernel_launch(void* const* d_in, const int* in_sizes, int n_in,
                              void* d_out, int out_size, void* d_ws, size_t ws_size,
                              hipStream_t stream) {
    const float* ts = (const float*)d_in[0];
    const float* nt = (const float*)d_in[1];
    float* lse_r = (float*)d_ws;            // 8192 floats
    float* lse_c = lse_r + N_TOTAL;         // 8192 floats (64 KB total)

    // Row-direction LSE: logits = 100 * ts @ nt^T
    lse_rows_kernel<<<N_TILES, 32, 0, stream>>>(ts, nt, lse_r);
    // Column-direction LSE: sim^T[i,j] = <nt_i, ts_j>; mask is symmetric.
    lse_rows_kernel<<<N_TILES, 32, 0, stream>>>(nt, ts, lse_c);

    finalize_kernel<<<1, 256, 0, stream>>>(ts, nt, lse_r, lse_c, (float*)d_out);
}